// Difference_78941498900584
// MI455X (gfx1250) — compile-verified
//
#include <hip/hip_runtime.h>

// Fused: P = relu(a @ feats^T), Q = (b @ feats^T <= 0) ? 1 : 0, out = P @ Q^T
// All GEMMs run on V_WMMA_F32_16X16X4_F32 (native fp32 matrix pipe, exact
// fp32 accumulate parity with the reference). Workspace use: 2 MB
// (P: 1024x256 f32, Q: 1024x256 f32).

typedef __attribute__((ext_vector_type(2))) float v2f;
typedef __attribute__((ext_vector_type(8))) float v8f;

#define D_DIM 512   // inner dim of stage-1 GEMMs
#define K_DIM 256   // feature count = inner dim of stage-2 GEMM
#define NA    1024
#define NB    1024

// ---------------------------------------------------------------------------
// Stage 1: rows 0..1023 = a, rows 1024..2047 = b. One 16x16 output tile per
// wave. A fragment: lane (l%16) = row, lanes>=16 take K offset +2 (f32 16x4
// layout). B[d][n] = feats[n][d] -> contiguous float2 per lane.
// ---------------------------------------------------------------------------
__global__ __launch_bounds__(256) void proj_kernel(const float* __restrict__ a,
                                                   const float* __restrict__ b,
                                                   const float* __restrict__ feats,
                                                   float* __restrict__ P,
                                                   float* __restrict__ Q) {
  const int lane  = threadIdx.x & 31;
  const int wave  = threadIdx.x >> 5;
  const int wid   = blockIdx.x * 8 + wave;       // 0 .. 2047
  const int tilesN = K_DIM / 16;                 // 16
  const int tileM  = wid / tilesN;               // 0 .. 127 (rows of concat(a,b))
  const int tileN  = wid % tilesN;
  const int mrow   = lane & 15;
  const int halfk  = lane >> 4;                  // 0: K{0,1}, 1: K{2,3}

  const bool isA     = (tileM < NA / 16);        // wave-uniform
  const float* X     = isA ? a : b;
  const int   rowTop = (isA ? tileM : tileM - NA / 16) * 16;

  const float* arow = X     + (size_t)(rowTop + mrow)      * D_DIM + halfk * 2;
  const float* brow = feats + (size_t)(tileN * 16 + mrow)  * D_DIM + halfk * 2;

  v8f acc = {};
#pragma unroll 8
  for (int kk = 0; kk < D_DIM; kk += 4) {
    v2f av = *(const v2f*)(arow + kk);
    v2f bv = *(const v2f*)(brow + kk);
    acc = __builtin_amdgcn_wmma_f32_16x16x4_f32(false, av, false, bv,
                                                (short)0, acc, false, false);
  }

  // C/D layout: VGPR r holds M = r + 8*halfk, N = mrow.
  float* Out = isA ? P : Q;
#pragma unroll
  for (int r = 0; r < 8; ++r) {
    const int m = rowTop + r + halfk * 8;
    float v = acc[r];
    v = isA ? fmaxf(v, 0.0f)                    // ReLU epilogue for P
            : ((v <= 0.0f) ? 1.0f : 0.0f);      // indicator epilogue for Q
    Out[(size_t)m * K_DIM + tileN * 16 + mrow] = v;
  }
}

// ---------------------------------------------------------------------------
// Stage 2: out(1024x1024) = P @ Q^T, K = 256. Each wave computes a 16x64
// strip: one A fragment reused across 4 B fragments (4 v_wmma per k-step).
// B[k][n] = Q[n][k] -> contiguous float2 per lane from row-major Q.
// ---------------------------------------------------------------------------
__global__ __launch_bounds__(256) void pair_kernel(const float* __restrict__ P,
                                                   const float* __restrict__ Q,
                                                   float* __restrict__ out) {
  const int lane   = threadIdx.x & 31;
  const int wave   = threadIdx.x >> 5;
  const int wid    = blockIdx.x * 8 + wave;      // 0 .. 1023
  const int groups = NB / 64;                    // 16 groups of 4 N-tiles
  const int tileM  = wid / groups;               // 0 .. 63
  const int ng     = wid % groups;
  const int mrow   = lane & 15;
  const int halfk  = lane >> 4;

  const float* arow = P + (size_t)(tileM * 16 + mrow) * K_DIM + halfk * 2;
  const float* br0  = Q + (size_t)((ng * 4 + 0) * 16 + mrow) * K_DIM + halfk * 2;
  const float* br1  = Q + (size_t)((ng * 4 + 1) * 16 + mrow) * K_DIM + halfk * 2;
  const float* br2  = Q + (size_t)((ng * 4 + 2) * 16 + mrow) * K_DIM + halfk * 2;
  const float* br3  = Q + (size_t)((ng * 4 + 3) * 16 + mrow) * K_DIM + halfk * 2;

  v8f acc0 = {}, acc1 = {}, acc2 = {}, acc3 = {};
#pragma unroll 4
  for (int kk = 0; kk < K_DIM; kk += 4) {
    v2f av = *(const v2f*)(arow + kk);
    v2f b0 = *(const v2f*)(br0 + kk);
    v2f b1 = *(const v2f*)(br1 + kk);
    v2f b2 = *(const v2f*)(br2 + kk);
    v2f b3 = *(const v2f*)(br3 + kk);
    acc0 = __builtin_amdgcn_wmma_f32_16x16x4_f32(false, av, false, b0,
                                                 (short)0, acc0, false, false);
    acc1 = __builtin_amdgcn_wmma_f32_16x16x4_f32(false, av, false, b1,
                                                 (short)0, acc1, false, false);
    acc2 = __builtin_amdgcn_wmma_f32_16x16x4_f32(false, av, false, b2,
                                                 (short)0, acc2, false, false);
    acc3 = __builtin_amdgcn_wmma_f32_16x16x4_f32(false, av, false, b3,
                                                 (short)0, acc3, false, false);
  }

#pragma unroll
  for (int r = 0; r < 8; ++r) {
    const int m = tileM * 16 + r + halfk * 8;
    float* orow = out + (size_t)m * NB + ng * 64 + mrow;
    orow[0]  = acc0[r];
    orow[16] = acc1[r];
    orow[32] = acc2[r];
    orow[48] = acc3[r];
  }
}

extern "C" void kernel_launch(void* const* d_in, const int* in_sizes, int n_in,
                              void* d_out, int out_size, void* d_ws, size_t ws_size,
                              hipStream_t stream) {
  const float* a     = (const float*)d_in[0];   // (1024, 512)
  const float* b     = (const float*)d_in[1];   // (1024, 512)
  const float* feats = (const float*)d_in[2];   // (256, 512)
  float* P   = (float*)d_ws;                    // 1024*256 f32
  float* Q   = P + (size_t)NA * K_DIM;          // 1024*256 f32 (2 MB total)
  float* out = (float*)d_out;                   // (1024, 1024)

  // Stage 1: 128 row-tiles x 16 col-tiles = 2048 waves -> 256 blocks of 8 waves
  proj_kernel<<<256, 256, 0, stream>>>(a, b, feats, P, Q);
  // Stage 2: 64 M-tiles x 16 N-groups = 1024 waves -> 128 blocks of 8 waves
  pair_kernel<<<128, 256, 0, stream>>>(P, Q, out);
}